// RGCNLayer_15444702396783
// MI455X (gfx1250) — compile-verified
//
#include <hip/hip_runtime.h>
#include <hip/hip_bf16.h>
#include <math.h>

typedef __attribute__((ext_vector_type(2))) float v2f;
typedef __attribute__((ext_vector_type(8))) float v8f;
typedef int v4i __attribute__((vector_size(16)));   // matches builtin param type

#define IN_F 128
#define HEADS 4
#define OUT_F 64
#define HD 256            // HEADS*OUT_F
#define WC 576            // 256 (W_fc) + 256 (gat_res_w) + 64 (res_w)
#define NEG_SLOPE 0.2f
#define BN_EPS 1e-5f

// ---------- helpers: monotonic float<->uint key for atomicMax over floats ----------
__device__ __forceinline__ unsigned fkey(float f) {
    unsigned b = __float_as_uint(f);
    return (b & 0x80000000u) ? ~b : (b | 0x80000000u);
}
__device__ __forceinline__ float funkey(unsigned k) {
    unsigned b = (k & 0x80000000u) ? (k & 0x7fffffffu) : ~k;
    return __uint_as_float(b);
}

// ---------- pack the three weight matrices into one [128 x 576] K-major matrix ----------
__global__ void pack_w(const float* __restrict__ Wfc, const float* __restrict__ Wres,
                       const float* __restrict__ Wr, float* __restrict__ Wcat) {
    int t = blockIdx.x * 256 + threadIdx.x;
    if (t >= IN_F * WC) return;
    int k = t / WC, c = t % WC;
    float v;
    if (c < 256)      v = Wfc[k * 256 + c];
    else if (c < 512) v = Wres[k * 256 + (c - 256)];
    else              v = Wr[k * 64 + (c - 512)];
    Wcat[t] = v;
}

// ---------- fused GEMM: Y[N x 576] = X[N x 128] @ Wcat[128 x 576], fp32 WMMA ----------
// block = 128 threads (4 waves); block tile = 16 rows x 64 cols; each wave owns a 16x16 tile.
__global__ __launch_bounds__(128) void gemm_wmma(const float* __restrict__ X,
                                                 const float* __restrict__ Wcat,
                                                 float* __restrict__ Y, int Nn) {
    __shared__ float Xs[16][IN_F];
    const int row0 = blockIdx.x * 16;
    const int wave = threadIdx.x >> 5;
    const int lane = threadIdx.x & 31;
    const bool full_tile = (row0 + 16 <= Nn);

    // stage the 16x128 fp32 A-tile in LDS
    if (full_tile) {
#if __has_builtin(__builtin_amdgcn_global_load_async_to_lds_b128)
        // CDNA5 async copy: 128 threads x 16B x 4 iters = 8KB, ASYNCcnt-tracked
        #pragma unroll
        for (int i = 0; i < 4; ++i) {
            int elt = i * 512 + threadIdx.x * 4;  // float index within tile
            const float* gp = X + (size_t)row0 * IN_F + elt;
            float* lp = &Xs[0][0] + elt;
            __builtin_amdgcn_global_load_async_to_lds_b128(
                (__attribute__((address_space(1))) v4i*)gp,
                (__attribute__((address_space(3))) v4i*)lp, 0, 0);
        }
#if __has_builtin(__builtin_amdgcn_s_wait_asynccnt)
        __builtin_amdgcn_s_wait_asynccnt(0);
#else
        asm volatile("s_wait_asynccnt 0" ::: "memory");
#endif
#else
        #pragma unroll
        for (int i = 0; i < 16; ++i)
            Xs[i][threadIdx.x] = X[(size_t)(row0 + i) * IN_F + threadIdx.x];
#endif
    } else {
        #pragma unroll
        for (int i = 0; i < 16; ++i) {
            int r = row0 + i;
            Xs[i][threadIdx.x] = (r < Nn) ? X[(size_t)r * IN_F + threadIdx.x] : 0.0f;
        }
    }
    __syncthreads();

    const int c0   = blockIdx.y * 64 + wave * 16;
    const int col  = c0 + (lane & 15);
    const int koff = (lane >> 4) * 2;   // lanes 0-15: K=0,1 ; lanes 16-31: K=2,3
    const int r    = lane & 15;

    v8f acc = {};
    #pragma unroll 4
    for (int kk = 0; kk < IN_F; kk += 4) {
        v2f a, b;
        a.x = Xs[r][kk + koff];
        a.y = Xs[r][kk + koff + 1];
        const float* wp = Wcat + (size_t)(kk + koff) * WC + col;
        b.x = wp[0];
        b.y = wp[WC];
        // D = A(16x4) * B(4x16) + C, fp32 matrix pipe
        acc = __builtin_amdgcn_wmma_f32_16x16x4_f32(
            /*neg_a=*/false, a, /*neg_b=*/false, b,
            /*c_mod=*/(short)0, acc, /*reuse_a=*/false, /*reuse_b=*/false);
    }

    // C/D layout: VGPR i -> lanes 0-15 row M=i, lanes 16-31 row M=8+i
    const int rbase = row0 + ((lane >> 4) * 8);
    if (full_tile) {
        float* yp = Y + (size_t)rbase * WC + col;
        #pragma unroll
        for (int i = 0; i < 8; ++i) yp[(size_t)i * WC] = acc[i];
    } else {
        #pragma unroll
        for (int i = 0; i < 8; ++i) {
            int rr = rbase + i;
            if (rr < Nn) Y[(size_t)rr * WC + col] = acc[i];
        }
    }
}

// ---------- per-node attention scores el/er: one wave per node, coalesced 1KB read ----------
__global__ __launch_bounds__(256) void attn_scores(const float* __restrict__ Y,
                                                   const float* __restrict__ al,
                                                   const float* __restrict__ ar,
                                                   float* __restrict__ el,
                                                   float* __restrict__ er, int Nn) {
    int wave = threadIdx.x >> 5;
    int lane = threadIdx.x & 31;
    int n = blockIdx.x * 8 + wave;
    if (n >= Nn) return;
    int h = lane >> 3;                     // lanes 8h..8h+7 cooperate on head h
    const float* hp  = Y + (size_t)n * WC + lane * 8;   // == h*64 + (lane&7)*8
    const float* alp = al + lane * 8;
    const float* arp = ar + lane * 8;
    float sl = 0.f, sr = 0.f;
    #pragma unroll
    for (int j = 0; j < 8; ++j) { float v = hp[j]; sl += v * alp[j]; sr += v * arp[j]; }
    sl += __shfl_down(sl, 4, 8);  sr += __shfl_down(sr, 4, 8);
    sl += __shfl_down(sl, 2, 8);  sr += __shfl_down(sr, 2, 8);
    sl += __shfl_down(sl, 1, 8);  sr += __shfl_down(sr, 1, 8);
    if ((lane & 7) == 0) { el[n * 4 + h] = sl; er[n * 4 + h] = sr; }
}

// ---------- edge scores: e = leakyrelu(el[src]+er[dst]); segment max via u32 keys ----------
__global__ void edge_scores(const float* __restrict__ el, const float* __restrict__ er,
                            const int* __restrict__ src, const int* __restrict__ dst,
                            float* __restrict__ ebuf, unsigned* __restrict__ mkeys, int E4) {
    int t = blockIdx.x * 256 + threadIdx.x;
    if (t >= E4) return;
    int e = t >> 2, h = t & 3;
    int s = src[e], d = dst[e];
    float v = el[s * 4 + h] + er[d * 4 + h];
    v = (v > 0.f) ? v : v * NEG_SLOPE;
    ebuf[t] = v;
    atomicMax(&mkeys[d * 4 + h], fkey(v));
}

// ---------- a = exp(e - m[dst]); segment sum ----------
__global__ void edge_exp(const int* __restrict__ dst, const unsigned* __restrict__ mkeys,
                         float* __restrict__ ebuf, float* __restrict__ ssum, int E4) {
    int t = blockIdx.x * 256 + threadIdx.x;
    if (t >= E4) return;
    int e = t >> 2, h = t & 3;
    int d = dst[e];
    float m = funkey(mkeys[d * 4 + h]);
    if (!__builtin_isfinite(m)) m = 0.f;   // empty-segment semantics of reference
    float a = __expf(ebuf[t] - m);
    ebuf[t] = a;                            // overwrite e with a (saves 12.8 MB scratch)
    atomicAdd(&ssum[d * 4 + h], a);
}

// ---------- rst[dst] += alpha * h[src]; one wave per edge, lane handles 8 features ----------
__global__ __launch_bounds__(256) void edge_aggregate(const float* __restrict__ Y,
                                                      const float* __restrict__ ebuf,
                                                      const float* __restrict__ ssum,
                                                      const int* __restrict__ src,
                                                      const int* __restrict__ dst,
                                                      float* __restrict__ rst, int E) {
    int wave = threadIdx.x >> 5;
    int lane = threadIdx.x & 31;
    int e = blockIdx.x * 8 + wave;
    if (e >= E) return;
    int s = src[e], d = dst[e];
    int h = lane >> 3;                      // lane*8 .. lane*8+7 stays within one head
    float alpha = ebuf[e * 4 + h] / (ssum[d * 4 + h] + 1e-16f);
    const float* hp = Y + (size_t)s * WC + lane * 8;   // h lives in Y[:, 0:256]
    float* rp = rst + (size_t)d * HD + lane * 8;
    #pragma unroll
    for (int j = 0; j < 8; ++j) atomicAdd(&rp[j], alpha * hp[j]);
}

// ---------- relu(rst + hres + bias), head-merge conv, outer residual, BN partial stats ----------
__global__ __launch_bounds__(512) void node_merge(const float* __restrict__ Y,
                                                  const float* __restrict__ rst,
                                                  const float* __restrict__ gbias,
                                                  const float* __restrict__ cw,
                                                  const float* __restrict__ cb,
                                                  const float* __restrict__ rb,
                                                  float* __restrict__ yout,
                                                  float* __restrict__ sums,
                                                  float* __restrict__ sumsq, int Nn) {
    __shared__ float s1[8][64];
    __shared__ float s2[8][64];
    int d  = threadIdx.x;                   // 0..63
    int ry = threadIdx.y;                   // 0..7
    int n  = blockIdx.x * 8 + ry;
    float v = 0.f;
    if (n < Nn) {
        #pragma unroll
        for (int h = 0; h < HEADS; ++h) {
            float t = rst[(size_t)n * HD + h * 64 + d]
                    + Y[(size_t)n * WC + 256 + h * 64 + d]
                    + gbias[h * 64 + d];
            v += cw[h] * fmaxf(t, 0.f);
        }
        v += cb[0];
        v += fmaxf(Y[(size_t)n * WC + 512 + d] + rb[d], 0.f);
        yout[(size_t)n * 64 + d] = v;       // staged pre-BN output in d_out
    }
    s1[ry][d] = (n < Nn) ? v : 0.f;
    s2[ry][d] = (n < Nn) ? v * v : 0.f;
    __syncthreads();
    if (ry == 0) {
        float a = 0.f, b = 0.f;
        #pragma unroll
        for (int i = 0; i < 8; ++i) { a += s1[i][d]; b += s2[i][d]; }
        atomicAdd(&sums[d], a);
        atomicAdd(&sumsq[d], b);
    }
}

// ---------- finalize BN stats into scale/shift ----------
__global__ void bn_stats(const float* __restrict__ sums, const float* __restrict__ sumsq,
                         const float* __restrict__ gamma, const float* __restrict__ beta,
                         float* __restrict__ ss, int Nn) {
    int d = threadIdx.x;
    if (d >= 64) return;
    float mean = sums[d] / (float)Nn;
    float var  = sumsq[d] / (float)Nn - mean * mean;
    float inv  = rsqrtf(var + BN_EPS);
    float sc   = inv * gamma[d];
    ss[d]      = sc;
    ss[64 + d] = beta[d] - mean * sc;
}

// ---------- apply BN in place on d_out ----------
__global__ void bn_apply(float* __restrict__ y, const float* __restrict__ ss, int total) {
    int t = blockIdx.x * 256 + threadIdx.x;
    if (t >= total) return;
    int d = t & 63;
    y[t] = y[t] * ss[d] + ss[64 + d];
}

extern "C" void kernel_launch(void* const* d_in, const int* in_sizes, int n_in,
                              void* d_out, int out_size, void* d_ws, size_t ws_size,
                              hipStream_t stream) {
    const float* X     = (const float*)d_in[0];
    const float* Wfc   = (const float*)d_in[1];
    const float* al    = (const float*)d_in[2];
    const float* ar    = (const float*)d_in[3];
    const float* Wres  = (const float*)d_in[4];
    const float* gbias = (const float*)d_in[5];
    const float* cw    = (const float*)d_in[6];
    const float* cb    = (const float*)d_in[7];
    const float* Wr    = (const float*)d_in[8];
    const float* rb    = (const float*)d_in[9];
    const float* gamma = (const float*)d_in[10];
    const float* beta  = (const float*)d_in[11];
    const int*   src   = (const int*)d_in[12];
    const int*   dst   = (const int*)d_in[13];
    const int Nn = in_sizes[0] / IN_F;
    const int E  = in_sizes[12];
    float* out = (float*)d_out;

    // ----- workspace carve: zero-initialized atomic region first (one memset) -----
    unsigned* mkeys = (unsigned*)d_ws;                 // N*4 (key 0 decodes non-finite -> handled)
    float* ssum  = (float*)(mkeys + (size_t)Nn * 4);   // N*4
    float* rst   = ssum + (size_t)Nn * 4;              // N*256
    float* sums  = rst + (size_t)Nn * HD;              // 64
    float* sumsq = sums + 64;                          // 64
    float* zend  = sumsq + 64;
    float* Y     = zend;                               // N*576
    float* el    = Y + (size_t)Nn * WC;                // N*4
    float* er    = el + (size_t)Nn * 4;                // N*4
    float* ebuf  = er + (size_t)Nn * 4;                // E*4 (e, then a in place)
    float* Wcat  = ebuf + (size_t)E * 4;               // 128*576
    float* ss    = Wcat + IN_F * WC;                   // 128 (scale|shift)

    size_t zbytes = (char*)zend - (char*)d_ws;
    (void)hipMemsetAsync(d_ws, 0, zbytes, stream);

    pack_w<<<(IN_F * WC + 255) / 256, 256, 0, stream>>>(Wfc, Wres, Wr, Wcat);

    dim3 ggrid((Nn + 15) / 16, WC / 64);
    gemm_wmma<<<ggrid, 128, 0, stream>>>(X, Wcat, Y, Nn);

    attn_scores<<<(Nn + 7) / 8, 256, 0, stream>>>(Y, al, ar, el, er, Nn);
    edge_scores<<<(E * HEADS + 255) / 256, 256, 0, stream>>>(el, er, src, dst, ebuf, mkeys, E * HEADS);
    edge_exp<<<(E * HEADS + 255) / 256, 256, 0, stream>>>(dst, mkeys, ebuf, ssum, E * HEADS);
    edge_aggregate<<<(E + 7) / 8, 256, 0, stream>>>(Y, ebuf, ssum, src, dst, rst, E);

    dim3 fb(64, 8);
    node_merge<<<(Nn + 7) / 8, fb, 0, stream>>>(Y, rst, gbias, cw, cb, rb, out, sums, sumsq, Nn);
    bn_stats<<<1, 64, 0, stream>>>(sums, sumsq, gamma, beta, ss, Nn);
    bn_apply<<<(Nn * 64 + 255) / 256, 256, 0, stream>>>(out, ss, Nn * 64);
}